// SputnikGATLayer_15324443312535
// MI455X (gfx1250) — compile-verified
//
#include <hip/hip_runtime.h>
#include <hip/hip_bf16.h>
#include <math.h>

// Problem constants (match reference)
#define NNODES   50000
#define DEG      16
#define IN_FEATS 256
#define HEADS    4
#define OUT_F    64
#define HF       (HEADS * OUT_F)   // 256
#define NEG_SLOPE 0.2f

typedef __attribute__((ext_vector_type(2))) float v2f;
typedef __attribute__((ext_vector_type(8))) float v8f;
typedef int v4i_b128 __attribute__((vector_size(16)));   // matches builtin param

#define AS_GLOBAL __attribute__((address_space(1)))
#define AS_LDS    __attribute__((address_space(3)))

#if defined(__has_builtin)
#if __has_builtin(__builtin_amdgcn_global_load_async_to_lds_b128)
#define HAVE_ASYNC_LDS 1
#endif
#endif

__device__ __forceinline__ void wait_asynccnt0() {
#if defined(__has_builtin) && __has_builtin(__builtin_amdgcn_s_wait_asynccnt)
    __builtin_amdgcn_s_wait_asynccnt(0);
#else
    asm volatile("s_wait_asynccnt 0x0" ::: "memory");
#endif
}

// ---------------------------------------------------------------------------
// Kernel 1: h = x @ W^T via V_WMMA_F32_16X16X4_F32
// grid = (NNODES/16, HF/128), block = 256 (8 wave32s)
// Block computes rows [bx*16, +16), cols [by*128 + wave*16, +16) per wave.
// A tile staged to LDS via GLOBAL_LOAD_ASYNC_TO_LDS_B128 when available
// (ASYNCcnt path, no VGPR round-trip), else cooperative VGPR staging.
// ---------------------------------------------------------------------------
__global__ __launch_bounds__(256) void gat_proj_wmma(const float* __restrict__ x,
                                                     const float* __restrict__ W,
                                                     float* __restrict__ h) {
    // 16 x 256 tile of x in LDS, padded stride 260 floats (row pitch 1040 B =
    // 65*16 B: keeps 16B alignment for async B128 copies AND rotates rows by
    // 4 banks each -> 16 rows cover all 64 banks -> conflict-free frag reads)
    constexpr int LS = 260;
    __shared__ float sx[16 * LS];

    const int row0 = blockIdx.x * 16;
    const int t    = threadIdx.x;

#if HAVE_ASYNC_LDS
    // 16 rows x 64 chunks of 16B = 1024 async B128 copies, 4 per thread
    for (int c = t; c < 16 * 64; c += 256) {
        const int r = c >> 6;
        const int o = (c & 63) * 4;   // float offset within the row
        __builtin_amdgcn_global_load_async_to_lds_b128(
            (AS_GLOBAL v4i_b128*)(x + (size_t)(row0 + r) * IN_FEATS + o),
            (AS_LDS v4i_b128*)(sx + r * LS + o),
            0, 0);
    }
    wait_asynccnt0();
#else
    // cooperative coalesced load of the A tile through VGPRs
    for (int i = t; i < 16 * IN_FEATS; i += 256) {
        const int r = i >> 8;
        const int c = i & 255;
        sx[r * LS + c] = x[(size_t)(row0 + r) * IN_FEATS + c];
    }
#endif
    __syncthreads();

    const int wave  = t >> 5;                 // 0..7
    const int lane  = t & 31;
    const int col0  = blockIdx.y * 128 + wave * 16;
    const int lrow  = lane & 15;              // M (A) / N (B,C)
    const int khalf = (lane >> 4) * 2;        // K sub-offset: 0 or 2

    v8f c = {};
    // B = W^T  =>  B[k][n] = W[n][k]; this lane's B column n = col0+lrow
    const float* wrow = W + (size_t)(col0 + lrow) * IN_FEATS;
    const float* arow = sx + lrow * LS + khalf;

    for (int k = 0; k < IN_FEATS; k += 4) {
        v2f a, b;
        a.x = arow[k];            // A[lrow][k + khalf]
        a.y = arow[k + 1];        // A[lrow][k + khalf + 1]
        b.x = wrow[k + khalf];    // B[k + khalf][col0+lrow]
        b.y = wrow[k + khalf + 1];
        // (neg_a, A, neg_b, B, c_mod, C, reuse_a, reuse_b)
        c = __builtin_amdgcn_wmma_f32_16x16x4_f32(
                false, a, false, b, (short)0, c, false, false);
    }

    // C/D layout: VGPR v, lanes 0-15 -> M=v, lanes 16-31 -> M=v+8; N = lane%16
    float* hp = h + (size_t)(row0 + 8 * (lane >> 4)) * HF + col0 + lrow;
#pragma unroll
    for (int v = 0; v < 8; ++v) {
        hp[(size_t)v * HF] = c[v];
    }
}

// ---------------------------------------------------------------------------
// Kernel 2: a_i[n,hd] = sum_f h[n,hd,f] * att_i[hd,f]  (and a_j likewise)
// one thread per (node, head)
// ---------------------------------------------------------------------------
__global__ __launch_bounds__(256) void gat_attn_halves(const float* __restrict__ h,
                                                       const float* __restrict__ att_i,
                                                       const float* __restrict__ att_j,
                                                       float* __restrict__ a_i,
                                                       float* __restrict__ a_j) {
    const int tg = blockIdx.x * blockDim.x + threadIdx.x;
    if (tg >= NNODES * HEADS) return;
    const int n  = tg >> 2;
    const int hd = tg & 3;
    const float* hp = h + (size_t)n * HF + hd * OUT_F;
    const float* wi = att_i + hd * OUT_F;
    const float* wj = att_j + hd * OUT_F;
    float si = 0.f, sj = 0.f;
#pragma unroll 8
    for (int f = 0; f < OUT_F; ++f) {
        const float v = hp[f];
        si += v * wi[f];
        sj += v * wj[f];
    }
    a_i[tg] = si;
    a_j[tg] = sj;
}

// ---------------------------------------------------------------------------
// Kernel 3: per-destination-node softmax over its 16 edges + weighted gather.
// One 256-thread block per node; thread tid owns output feature tid
// (head = tid/64). Neighbor rows are contiguous 1KB reads (L2-resident: h is
// 51MB, L2 is 192MB).
// ---------------------------------------------------------------------------
__global__ __launch_bounds__(256) void gat_aggregate(const float* __restrict__ h,
                                                     const int* __restrict__ col_id,
                                                     const float* __restrict__ a_i,
                                                     const float* __restrict__ a_j,
                                                     float* __restrict__ out) {
    __shared__ int   s_col[DEG];
    __shared__ float s_e[HEADS][DEG];
    __shared__ float s_alpha[HEADS][DEG];

    const int n = blockIdx.x;
    const int t = threadIdx.x;

    if (t < DEG) s_col[t] = col_id[n * DEG + t];
    __syncthreads();

    // leaky-relu edge logits: 64 threads = (head, k)
    if (t < HEADS * DEG) {
        const int k  = t & (DEG - 1);
        const int hd = t >> 4;
        float e = a_i[n * HEADS + hd] + a_j[s_col[k] * HEADS + hd];
        e = (e > 0.f) ? e : NEG_SLOPE * e;
        s_e[hd][k] = e;
    }
    __syncthreads();

    // stable softmax over 16 edges, one thread per head
    if (t < HEADS) {
        float m = -INFINITY;
#pragma unroll
        for (int k = 0; k < DEG; ++k) m = fmaxf(m, s_e[t][k]);
        if (!isfinite(m)) m = 0.f;
        float s = 0.f;
#pragma unroll
        for (int k = 0; k < DEG; ++k) {
            const float p = __expf(s_e[t][k] - m);
            s_alpha[t][k] = p;
            s += p;
        }
        const float inv = 1.f / (s + 1e-16f);
#pragma unroll
        for (int k = 0; k < DEG; ++k) s_alpha[t][k] *= inv;
    }
    __syncthreads();

    // weighted gather-sum over 16 neighbor rows
    const int hd = t >> 6;
    float acc = 0.f;
#pragma unroll 4
    for (int k = 0; k < DEG; ++k) {
        if (k + 1 < DEG) {
            __builtin_prefetch(h + (size_t)s_col[k + 1] * HF + t, 0, 0);
        }
        acc += s_alpha[hd][k] * h[(size_t)s_col[k] * HF + t];
    }
    out[(size_t)n * HF + t] = acc;
}

// ---------------------------------------------------------------------------
// launcher
// d_in order: x, row_id, row_ptr, col_id, W, att_i, att_j
// workspace layout: h [N*256] | a_i [N*4] | a_j [N*4]   (all f32)
// ---------------------------------------------------------------------------
extern "C" void kernel_launch(void* const* d_in, const int* in_sizes, int n_in,
                              void* d_out, int out_size, void* d_ws, size_t ws_size,
                              hipStream_t stream) {
    const float* x      = (const float*)d_in[0];
    // d_in[1] = row_id, d_in[2] = row_ptr: implicit with fixed degree, unused
    const int*   col_id = (const int*)d_in[3];
    const float* W      = (const float*)d_in[4];
    const float* att_i  = (const float*)d_in[5];
    const float* att_j  = (const float*)d_in[6];
    float*       out    = (float*)d_out;

    float* h   = (float*)d_ws;                       // N * 256
    float* a_i = h + (size_t)NNODES * HF;            // N * 4
    float* a_j = a_i + (size_t)NNODES * HEADS;       // N * 4

    dim3 g1(NNODES / 16, HF / 128);                  // 3125 x 2
    gat_proj_wmma<<<g1, 256, 0, stream>>>(x, W, h);

    const int tot = NNODES * HEADS;
    gat_attn_halves<<<(tot + 255) / 256, 256, 0, stream>>>(h, att_i, att_j, a_i, a_j);

    gat_aggregate<<<NNODES, 256, 0, stream>>>(h, col_id, a_i, a_j, out);
}